// linearized_transform_44057774523007
// MI455X (gfx1250) — compile-verified
//
#include <hip/hip_runtime.h>

// CDNA5 / gfx1250, wave32. One wave computes one 16x16 output tile.
// Affine grid via V_WMMA_F32_16X16X4_F32; bilinear gather per lane.

typedef __attribute__((ext_vector_type(2))) float v2f;
typedef __attribute__((ext_vector_type(8))) float v8f;

#define IMG_H   1024
#define IMG_W   1024
#define CHANS   3
#define BATCH   4
#define HW      (IMG_H * IMG_W)           // 1048576
#define TILES_X (IMG_W / 16)              // 64
#define TILES_Y (IMG_H / 16)              // 64
#define TILES_PER_B (TILES_X * TILES_Y)   // 4096
#define TOTAL_TILES (TILES_PER_B * BATCH) // 16384
#define WAVES_PER_BLOCK 8
#define NUM_BLOCKS (TOTAL_TILES / WAVES_PER_BLOCK) // 2048

__global__ __launch_bounds__(256)
void lin_grid_sample_kernel(const float* __restrict__ image,
                            const float* __restrict__ theta,
                            float* __restrict__ out)
{
    const int lane = threadIdx.x & 31;
    const int wave = threadIdx.x >> 5;
    const int tile = blockIdx.x * WAVES_PER_BLOCK + wave;

    const int b   = tile >> 12;        // / TILES_PER_B
    const int trm = tile & 4095;
    const int ty  = trm >> 6;          // / TILES_X
    const int tx  = trm & 63;
    const int h0  = ty << 4;
    const int w0  = tx << 4;

    // theta[b][2][3]
    const float t00 = theta[b * 6 + 0];
    const float t01 = theta[b * 6 + 1];
    const float t02 = theta[b * 6 + 2];
    const float t10 = theta[b * 6 + 3];
    const float t11 = theta[b * 6 + 4];
    const float t12 = theta[b * 6 + 5];

    const bool  lo  = (lane < 16);
    const int   l16 = lane & 15;
    const float inv = 2.0f / 1023.0f;  // align_corners linspace step
    const float Ym  = fmaf((float)(h0 + l16), inv, -1.0f); // row coord (A, lanes 0-15)
    const float Xn  = fmaf((float)(w0 + l16), inv, -1.0f); // col coord (B, lanes 0-15)

    // A (16x4): lanes0-15 {K0=Y_m, K1=1}; lanes16-31 {K2=0, K3=0}
    v2f a;
    a.x = lo ? Ym   : 0.0f;
    a.y = lo ? 1.0f : 0.0f;
    // B (4x16): row K0 = t01 (Y coefficient); row K1 = t00*X_n + t02; K2=K3=0
    v2f bx, by;
    bx.x = lo ? t01 : 0.0f;
    bx.y = lo ? fmaf(t00, Xn, t02) : 0.0f;
    by.x = lo ? t11 : 0.0f;
    by.y = lo ? fmaf(t10, Xn, t12) : 0.0f;

    v8f czero = {};
    // D[m][n] = grid_x / grid_y for pixel (h0+m, w0+n)
    v8f gx = __builtin_amdgcn_wmma_f32_16x16x4_f32(
        false, a, false, bx, (short)0, czero, false, false);
    v8f gy = __builtin_amdgcn_wmma_f32_16x16x4_f32(
        false, a, false, by, (short)0, czero, false, false);

    const float* imgB = image + (size_t)b * (CHANS * HW);
    float*       outB = out   + (size_t)b * (CHANS * HW);

    const int m_add = lo ? 0 : 8;
    const int n     = l16;

#pragma unroll
    for (int v = 0; v < 8; ++v) {
        const float gxv = gx[v];
        const float gyv = gy[v];
        const int h = h0 + v + m_add;
        const int w = w0 + n;

        // align_corners unnormalize
        const float ix = (gxv + 1.0f) * 0.5f * (float)(IMG_W - 1);
        const float iy = (gyv + 1.0f) * 0.5f * (float)(IMG_H - 1);
        const float x0f = floorf(ix);
        const float y0f = floorf(iy);
        const float wx1 = ix - x0f;
        const float wy1 = iy - y0f;
        const float wx0 = 1.0f - wx1;
        const float wy0 = 1.0f - wy1;
        const int x0 = (int)x0f;
        const int y0 = (int)y0f;

        float acc0 = 0.0f, acc1 = 0.0f, acc2 = 0.0f;
#pragma unroll
        for (int dy = 0; dy < 2; ++dy) {
#pragma unroll
            for (int dx = 0; dx < 2; ++dx) {
                const int xi = x0 + dx;
                const int yi = y0 + dy;
                const bool valid = (xi >= 0) && (xi <= IMG_W - 1) &&
                                   (yi >= 0) && (yi <= IMG_H - 1);
                const int xc = min(max(xi, 0), IMG_W - 1);
                const int yc = min(max(yi, 0), IMG_H - 1);
                const float wgt = (dx ? wx1 : wx0) * (dy ? wy1 : wy0) *
                                  (valid ? 1.0f : 0.0f);
                const size_t off = (size_t)yc * IMG_W + xc;
                acc0 = fmaf(imgB[off],            wgt, acc0);
                acc1 = fmaf(imgB[off + HW],       wgt, acc1);
                acc2 = fmaf(imgB[off + 2 * HW],   wgt, acc2);
            }
        }
        const size_t o = (size_t)h * IMG_W + w;
        outB[o]          = acc0;
        outB[o + HW]     = acc1;
        outB[o + 2 * HW] = acc2;
    }
}

extern "C" void kernel_launch(void* const* d_in, const int* in_sizes, int n_in,
                              void* d_out, int out_size, void* d_ws, size_t ws_size,
                              hipStream_t stream) {
    (void)in_sizes; (void)n_in; (void)out_size; (void)d_ws; (void)ws_size;
    const float* image = (const float*)d_in[0];  // (4,3,1024,1024) f32
    const float* theta = (const float*)d_in[1];  // (4,2,3) f32
    float* out = (float*)d_out;                  // (4,3,1024,1024) f32

    lin_grid_sample_kernel<<<NUM_BLOCKS, 256, 0, stream>>>(image, theta, out);
}